// BayesianLayer_50611894616336
// MI455X (gfx1250) — compile-verified
//
#include <hip/hip_runtime.h>
#include <hip/hip_bf16.h>
#include <math.h>

// Problem dims (fixed by reference)
#define B_   32
#define R_   4
#define IN_  1024
#define OUT_ 1024
#define ISPLIT 8                 // i-range splits per (b,r) in the streaming pass
#define IPER  (IN_ / ISPLIT)     // 128 rows per block

typedef __attribute__((ext_vector_type(2))) float v2f;
typedef __attribute__((ext_vector_type(8))) float v8f;

// ---- workspace layout (float offsets) ----
#define WS_SCALARS 0                         // [0]=prior_sum  [1]=logp_sum (atomic accum)
#define WS_OUTACC  16                        // B*OUT : sum_r  x . (eps*sigma)
#define WS_BIASSUM (16 + B_*OUT_)            // B*OUT : sum_r bias
#define WS_GEMM    (WS_BIASSUM + B_*OUT_)    // B*OUT : x @ mu  (WMMA)
#define WS_SIGMA   (WS_GEMM + B_*OUT_)       // IN*OUT : softplus(ro)
#define WS_LOGSIG  (WS_SIGMA + IN_*OUT_)     // IN*OUT : log(sigma)

#define INV_SQRT_2PI 0.3989422804014327f     // 1/sqrt(2pi)          (sigma1=1.0)
#define INV_SQRT_2PI_01 3.989422804014327f   // 1/(sqrt(2pi)*0.1)    (sigma2=0.1)
#define HALF_LOG_2PI 0.9189385332046727f
#define LOG_CLIP_LO -23.025850929940457f     // log(1e-10)
#define LOG_CLIP_HI  2.302585092994046f      // log(10)

__device__ __forceinline__ float clampf(float v, float lo, float hi) {
    return fminf(fmaxf(v, lo), hi);
}

// -------- zero the atomic accumulators (scalars + outacc) each launch --------
__global__ void k_init(float* ws) {
    int i = blockIdx.x * 256 + threadIdx.x;
    if (i < WS_BIASSUM) ws[i] = 0.0f;
}

// -------- sigma = softplus(ro), log(sigma) : computed once, reused 128x ------
__global__ void k_prep(const float* __restrict__ ro, float* __restrict__ sigma,
                       float* __restrict__ logsig) {
    int i = blockIdx.x * 256 + threadIdx.x;   // < IN_*OUT_
    float v = ro[i];
    float s = log1pf(__expf(v));
    sigma[i]  = s;
    logsig[i] = __logf(s);
}

// -------- bias pass: bias sums + bias log-terms (tiny: B*R*OUT elems) --------
__global__ void k_bias(const float* __restrict__ mu_bias, const float* __restrict__ ro_bias,
                       const float* __restrict__ eps_bias, float* ws) {
    float* scal = ws + WS_SCALARS;
    float* bsum = ws + WS_BIASSUM;
    const int tid = threadIdx.x;
    const int b = blockIdx.y;
    const int o = blockIdx.x * 256 + tid;

    float rb  = ro_bias[o];
    float sb  = log1pf(__expf(rb));
    float lsb = __logf(sb);
    float mb  = mu_bias[o];

    float bs = 0.f, sprior = 0.f, slogp = 0.f;
    for (int r = 0; r < R_; ++r) {
        float e = eps_bias[(b * R_ + r) * OUT_ + o];
        float bias = fmaf(e, sb, mb);
        bs += bias;
        float g1 = clampf(INV_SQRT_2PI    * __expf(-0.5f * bias * bias), 1e-10f, 10.f);
        float g2 = clampf(INV_SQRT_2PI_01 * __expf(-50.f * bias * bias), 1e-10f, 10.f);
        sprior += __logf(0.5f * (g1 + g2));
        // log N(bias; mu_b, sb): z = eps ; clip in log-domain (monotone)
        slogp += clampf(fmaf(-0.5f * e, e, -lsb) - HALF_LOG_2PI, LOG_CLIP_LO, LOG_CLIP_HI);
    }
    bsum[b * OUT_ + o] = bs;

    __shared__ float red[256];
    red[tid] = sprior; __syncthreads();
    for (int s = 128; s > 0; s >>= 1) { if (tid < s) red[tid] += red[tid + s]; __syncthreads(); }
    if (tid == 0) atomicAdd(&scal[0], red[0]);
    __syncthreads();
    red[tid] = slogp; __syncthreads();
    for (int s = 128; s > 0; s >>= 1) { if (tid < s) red[tid] += red[tid + s]; __syncthreads(); }
    if (tid == 0) atomicAdd(&scal[1], red[0]);
}

// -------- main streaming pass over eps (512 MB, HBM-bound) -------------------
// grid: (ISPLIT, R, B); block 256. Each thread owns 4 consecutive columns
// (float4 / b128 loads: 4x fewer vmem instructions), the block covers all 1024
// columns, and marches down its 128-row i-slice. eps element read exactly once;
// sigma/mu/logsig re-reads served from the 192MB L2.
__global__ void k_main(const float* __restrict__ x, const float* __restrict__ mu,
                       const float* __restrict__ eps, float* ws) {
    const float4* sigma4  = (const float4*)(ws + WS_SIGMA);
    const float4* logsig4 = (const float4*)(ws + WS_LOGSIG);
    const float4* mu4     = (const float4*)mu;
    float* outacc = ws + WS_OUTACC;
    float* scal   = ws + WS_SCALARS;

    const int tid = threadIdx.x;
    const int o4  = tid;                    // float4 column index: columns 4*tid..4*tid+3
    const int i0  = blockIdx.x * IPER;
    const int r   = blockIdx.y;
    const int b   = blockIdx.z;

    __shared__ float xs[IPER];
    if (tid < IPER) xs[tid] = x[b * IN_ + i0 + tid];
    __syncthreads();

    const float4* ep4 = (const float4*)(eps + ((size_t)(b * R_ + r) * IN_ + i0) * OUT_) + o4;
    const int rowstride4 = OUT_ / 4;        // float4 stride between i rows

    float4 acc = make_float4(0.f, 0.f, 0.f, 0.f);
    float sprior = 0.f, slogp = 0.f;

    for (int i = 0; i < IPER; ++i) {
        float4 e  = ep4[(size_t)i * rowstride4];
        int ip = (i + 16 < IPER) ? (i + 16) : (IPER - 1);
        __builtin_prefetch(ep4 + (size_t)ip * rowstride4, 0, 0);  // global_prefetch_b8
        float4 sg = sigma4 [(i0 + i) * rowstride4 + o4];
        float4 m  = mu4    [(i0 + i) * rowstride4 + o4];
        float4 ls = logsig4[(i0 + i) * rowstride4 + o4];
        float xi  = xs[i];

        float ec[4] = {e.x, e.y, e.z, e.w};
        float sc[4] = {sg.x, sg.y, sg.z, sg.w};
        float mc[4] = {m.x, m.y, m.z, m.w};
        float lc[4] = {ls.x, ls.y, ls.z, ls.w};
        float ac[4] = {acc.x, acc.y, acc.z, acc.w};
#pragma unroll
        for (int c = 0; c < 4; ++c) {
            float t = ec[c] * sc[c];        // eps*sigma
            float w = t + mc[c];            // weight sample
            ac[c] = fmaf(xi, t, ac[c]);     // x . (eps*sigma); x@mu added via WMMA
            // log prior: mixture of two zero-mean gaussians, pdf clipped then logged
            float g1 = clampf(INV_SQRT_2PI    * __expf(-0.5f * w * w), 1e-10f, 10.f);
            float g2 = clampf(INV_SQRT_2PI_01 * __expf(-50.f * w * w), 1e-10f, 10.f);
            sprior += __logf(0.5f * (g1 + g2));
            // log q(w|mu,sigma): z-score is exactly eps; clip in log-domain (monotone)
            slogp += clampf(fmaf(-0.5f * ec[c], ec[c], -lc[c]) - HALF_LOG_2PI,
                            LOG_CLIP_LO, LOG_CLIP_HI);
        }
        acc = make_float4(ac[0], ac[1], ac[2], ac[3]);
    }

    float* oa = &outacc[b * OUT_ + 4 * o4];
    atomicAdd(oa + 0, acc.x);
    atomicAdd(oa + 1, acc.y);
    atomicAdd(oa + 2, acc.z);
    atomicAdd(oa + 3, acc.w);

    __shared__ float red[256];
    red[tid] = sprior; __syncthreads();
    for (int s = 128; s > 0; s >>= 1) { if (tid < s) red[tid] += red[tid + s]; __syncthreads(); }
    if (tid == 0) atomicAdd(&scal[0], red[0]);
    __syncthreads();
    red[tid] = slogp; __syncthreads();
    for (int s = 128; s > 0; s >>= 1) { if (tid < s) red[tid] += red[tid + s]; __syncthreads(); }
    if (tid == 0) atomicAdd(&scal[1], red[0]);
}

// -------- x @ mu via V_WMMA_F32_16X16X4_F32 (full f32, shared across r) ------
// grid (2, 64), block 32 (one wave). A = x tile (16x4), B = mu tile (4x16).
// Layouts per ISA 7.12.2: A lane<16 -> K=0,1 ; lane>=16 -> K=2,3.
//                         C/D: VGPR v -> M=v (lanes 0-15) / M=v+8 (lanes 16-31).
__global__ void k_gemm(const float* __restrict__ x, const float* __restrict__ mu, float* ws) {
    float* g = ws + WS_GEMM;
    const int lane = threadIdx.x;       // 0..31
    const int half = lane >> 4;         // 0 | 1
    const int ml   = lane & 15;
    const int m0   = blockIdx.x * 16;   // 0 or 16  (batch rows)
    const int n0   = blockIdx.y * 16;   // output column tile

    v8f c = {0.f, 0.f, 0.f, 0.f, 0.f, 0.f, 0.f, 0.f};
    const float* arow = x + (m0 + ml) * IN_;

    for (int k = 0; k < IN_; k += 4) {
        int ka = k + 2 * half;
        v2f a, bb;
        a.x  = arow[ka];
        a.y  = arow[ka + 1];
        bb.x = mu[(size_t)ka       * OUT_ + n0 + ml];
        bb.y = mu[(size_t)(ka + 1) * OUT_ + n0 + ml];
        c = __builtin_amdgcn_wmma_f32_16x16x4_f32(false, a, false, bb,
                                                  (short)0, c, false, false);
    }
#pragma unroll
    for (int v = 0; v < 8; ++v)
        g[(size_t)(m0 + v + 8 * half) * OUT_ + n0 + ml] = c[v];
}

// -------- finalize: out = x@mu + (sum_r part + sum_r bias)/R ; scalars /128 --
__global__ void k_final(float* __restrict__ out, const float* __restrict__ ws) {
    int i = blockIdx.x * 256 + threadIdx.x;     // < B*OUT
    out[i] = ws[WS_GEMM + i] + 0.25f * (ws[WS_OUTACC + i] + ws[WS_BIASSUM + i]);
    if (i == 0) {
        out[B_ * OUT_]     = ws[0] * (1.0f / 128.0f);   // log_prior
        out[B_ * OUT_ + 1] = ws[1] * (1.0f / 128.0f);   // log_p_weights
    }
}

extern "C" void kernel_launch(void* const* d_in, const int* in_sizes, int n_in,
                              void* d_out, int out_size, void* d_ws, size_t ws_size,
                              hipStream_t stream) {
    const float* x        = (const float*)d_in[0];
    const float* mu       = (const float*)d_in[1];
    const float* ro       = (const float*)d_in[2];
    const float* mu_bias  = (const float*)d_in[3];
    const float* ro_bias  = (const float*)d_in[4];
    const float* eps      = (const float*)d_in[5];
    const float* eps_bias = (const float*)d_in[6];
    float* out = (float*)d_out;
    float* ws  = (float*)d_ws;

    k_init <<<(WS_BIASSUM + 255) / 256, 256, 0, stream>>>(ws);
    k_prep <<<(IN_ * OUT_) / 256,       256, 0, stream>>>(ro, ws + WS_SIGMA, ws + WS_LOGSIG);
    k_bias <<<dim3(OUT_ / 256, B_),     256, 0, stream>>>(mu_bias, ro_bias, eps_bias, ws);
    k_main <<<dim3(ISPLIT, R_, B_),     256, 0, stream>>>(x, mu, eps, ws);
    k_gemm <<<dim3(B_ / 16, OUT_ / 16),  32, 0, stream>>>(x, mu, ws);
    k_final<<<(B_ * OUT_) / 256,        256, 0, stream>>>(out, ws);
}